// GCLEncoder_33191507264214
// MI455X (gfx1250) — compile-verified
//
#include <hip/hip_runtime.h>

typedef __attribute__((ext_vector_type(2))) float v2f;
typedef __attribute__((ext_vector_type(8))) float v8f;

#define IN_DIM  128
#define HIDDEN  64
#define OUT_DIM 32

// ---------------- degree / normalization ----------------

__global__ void k_init_deg(float* __restrict__ deg, int n) {
  int i = blockIdx.x * blockDim.x + threadIdx.x;
  if (i < n) deg[i] = 1.0f;   // self-loop contributes 1 to every node's degree
}

__global__ void k_accum_deg(const int* __restrict__ dst, float* __restrict__ deg, int e) {
  int i = blockIdx.x * blockDim.x + threadIdx.x;
  if (i < e) atomicAdd(&deg[dst[i]], 1.0f);
}

__global__ void k_dinv(float* __restrict__ deg, int n) {
  int i = blockIdx.x * blockDim.x + threadIdx.x;
  if (i < n) {
    float d = deg[i];
    deg[i] = (d > 0.0f) ? rsqrtf(d) : 0.0f;
  }
}

// ---------------- GEMM1: h1[M,64] = x[M,128] @ W1[128,64]  (WMMA f32 16x16x4) ----------------

__global__ void k_gemm1(const float* __restrict__ x, const float* __restrict__ W1,
                        float* __restrict__ h1, int mtiles) {
  const int wave  = blockIdx.x * (blockDim.x >> 5) + (threadIdx.x >> 5);
  const int nwaves = mtiles * (HIDDEN / 16);
  if (wave >= nwaves) return;              // uniform per-wave: EXEC stays all-1s for WMMA
  const int lane  = threadIdx.x & 31;
  const int mt    = wave >> 2;             // HIDDEN/16 == 4 tiles in N
  const int nt    = wave & 3;
  const int m0    = mt * 16;
  const int n0    = nt * 16;
  const int lhalf = lane >> 4;             // 0: lanes 0-15, 1: lanes 16-31
  const int l15   = lane & 15;
  const int koff  = lhalf * 2;             // A/B K sub-offset per ISA layout

  v8f acc = {};
  const float* arow = x  + (size_t)(m0 + l15) * IN_DIM + koff;   // A: row m0+l15
  const float* bcol = W1 + (size_t)koff * HIDDEN + n0 + l15;     // B: col n0+l15

  #pragma unroll 4
  for (int k = 0; k < IN_DIM; k += 4) {
    v2f a, b;
    a.x = arow[k];
    a.y = arow[k + 1];
    b.x = bcol[(size_t)k * HIDDEN];
    b.y = bcol[(size_t)(k + 1) * HIDDEN];
    acc = __builtin_amdgcn_wmma_f32_16x16x4_f32(false, a, false, b,
                                                (short)0, acc, false, false);
  }

  #pragma unroll
  for (int v = 0; v < 8; ++v) {
    int row = m0 + v + lhalf * 8;          // C/D layout: VGPR v -> M=v (+8 for hi lanes)
    h1[(size_t)row * HIDDEN + n0 + l15] = acc[v];
  }
}

// ---------------- layer-1 aggregation ----------------

// self-loop term: agg1[i][c] = h1[i][c] * dinv[i]^2
__global__ void k_selfloop1(const float* __restrict__ h1, const float* __restrict__ dinv,
                            float* __restrict__ agg1, int total) {
  int idx = blockIdx.x * blockDim.x + threadIdx.x;
  if (idx >= total) return;
  int i = idx >> 6;                        // HIDDEN = 64
  float di = dinv[i];
  agg1[idx] = h1[idx] * di * di;
}

// edge scatter: one wave per edge, 2 channels per lane (64 channels)
__global__ void k_scatter1(const int* __restrict__ src, const int* __restrict__ dst,
                           const float* __restrict__ dinv, const float* __restrict__ h1,
                           float* __restrict__ agg1, int e) {
  int idx = blockIdx.x * blockDim.x + threadIdx.x;
  int edge = idx >> 5;
  if (edge >= e) return;
  int lane = idx & 31;
  int s = src[edge];
  int d = dst[edge];
  float norm = dinv[s] * dinv[d];
  int c = lane * 2;
  const float2 hv = *(const float2*)(h1 + (size_t)s * HIDDEN + c);
  float* out = agg1 + (size_t)d * HIDDEN + c;
  atomicAdd(out,     hv.x * norm);
  atomicAdd(out + 1, hv.y * norm);
}

// ---------------- GEMM2: z[M,32] = relu(agg1+b1)[M,64] @ W2[64,32] (fused bias+ReLU) ----------------

__global__ void k_gemm2(const float* __restrict__ agg1, const float* __restrict__ b1,
                        const float* __restrict__ W2, float* __restrict__ z, int mtiles) {
  const int wave  = blockIdx.x * (blockDim.x >> 5) + (threadIdx.x >> 5);
  const int nwaves = mtiles * (OUT_DIM / 16);
  if (wave >= nwaves) return;
  const int lane  = threadIdx.x & 31;
  const int mt    = wave >> 1;             // OUT_DIM/16 == 2 tiles in N
  const int nt    = wave & 1;
  const int m0    = mt * 16;
  const int n0    = nt * 16;
  const int lhalf = lane >> 4;
  const int l15   = lane & 15;
  const int koff  = lhalf * 2;

  v8f acc = {};
  const float* arow = agg1 + (size_t)(m0 + l15) * HIDDEN + koff;
  const float* bcol = W2   + (size_t)koff * OUT_DIM + n0 + l15;

  #pragma unroll 4
  for (int k = 0; k < HIDDEN; k += 4) {
    v2f a, b;
    float a0 = arow[k]     + b1[k + koff];
    float a1 = arow[k + 1] + b1[k + koff + 1];
    a.x = a0 > 0.0f ? a0 : 0.0f;           // fused bias + ReLU
    a.y = a1 > 0.0f ? a1 : 0.0f;
    b.x = bcol[(size_t)k * OUT_DIM];
    b.y = bcol[(size_t)(k + 1) * OUT_DIM];
    acc = __builtin_amdgcn_wmma_f32_16x16x4_f32(false, a, false, b,
                                                (short)0, acc, false, false);
  }

  #pragma unroll
  for (int v = 0; v < 8; ++v) {
    int row = m0 + v + lhalf * 8;
    z[(size_t)row * OUT_DIM + n0 + l15] = acc[v];
  }
}

// ---------------- layer-2 aggregation ----------------

// out[i][c] = z[i][c]*dinv[i]^2 + b2[c]   (self-loop + bias; scatter adds on top)
__global__ void k_initout(const float* __restrict__ z, const float* __restrict__ dinv,
                          const float* __restrict__ b2, float* __restrict__ out, int total) {
  int idx = blockIdx.x * blockDim.x + threadIdx.x;
  if (idx >= total) return;
  int i = idx >> 5;                        // OUT_DIM = 32
  int c = idx & 31;
  float di = dinv[i];
  out[idx] = z[idx] * di * di + b2[c];
}

// edge scatter: one wave per edge, 1 channel per lane (32 channels)
__global__ void k_scatter2(const int* __restrict__ src, const int* __restrict__ dst,
                           const float* __restrict__ dinv, const float* __restrict__ z,
                           float* __restrict__ out, int e) {
  int idx = blockIdx.x * blockDim.x + threadIdx.x;
  int edge = idx >> 5;
  if (edge >= e) return;
  int lane = idx & 31;
  int s = src[edge];
  int d = dst[edge];
  float norm = dinv[s] * dinv[d];
  float v = z[(size_t)s * OUT_DIM + lane] * norm;
  atomicAdd(out + (size_t)d * OUT_DIM + lane, v);
}

// ---------------- launch ----------------

extern "C" void kernel_launch(void* const* d_in, const int* in_sizes, int n_in,
                              void* d_out, int out_size, void* d_ws, size_t ws_size,
                              hipStream_t stream) {
  const float* x  = (const float*)d_in[0];
  const float* W1 = (const float*)d_in[1];
  const float* b1 = (const float*)d_in[2];
  const float* W2 = (const float*)d_in[3];
  const float* b2 = (const float*)d_in[4];
  const int* ei   = (const int*)d_in[5];

  const int N = in_sizes[0] / IN_DIM;        // 100000 (divisible by 16)
  const int E = in_sizes[5] / 2;             // 1600000
  const int* src = ei;
  const int* dst = ei + E;

  float* ws   = (float*)d_ws;
  float* dinv = ws;                          // N
  float* h1z  = ws + N;                      // N*64 (h1 for layer1, reused as z for layer2)
  float* agg1 = ws + N + (size_t)N * HIDDEN; // N*64
  float* out  = (float*)d_out;               // N*32

  const int B = 256;

  // normalization
  k_init_deg<<<(N + B - 1) / B, B, 0, stream>>>(dinv, N);
  k_accum_deg<<<(E + B - 1) / B, B, 0, stream>>>(dst, dinv, E);
  k_dinv<<<(N + B - 1) / B, B, 0, stream>>>(dinv, N);

  // layer 1
  const int mtiles = N / 16;
  {
    int waves = mtiles * (HIDDEN / 16);
    int blocks = (waves + (B / 32) - 1) / (B / 32);
    k_gemm1<<<blocks, B, 0, stream>>>(x, W1, h1z, mtiles);
  }
  {
    int total = N * HIDDEN;
    k_selfloop1<<<(total + B - 1) / B, B, 0, stream>>>(h1z, dinv, agg1, total);
    int sthreads = E * 32;
    k_scatter1<<<(sthreads + B - 1) / B, B, 0, stream>>>(src, dst, dinv, h1z, agg1, E);
  }

  // layer 2 (bias+ReLU fused into GEMM2 A-load); z reuses h1 storage
  {
    int waves = mtiles * (OUT_DIM / 16);
    int blocks = (waves + (B / 32) - 1) / (B / 32);
    k_gemm2<<<blocks, B, 0, stream>>>(agg1, b1, W2, h1z, mtiles);
  }
  {
    int total = N * OUT_DIM;
    k_initout<<<(total + B - 1) / B, B, 0, stream>>>(h1z, dinv, b2, out, total);
    int sthreads = E * 32;
    k_scatter2<<<(sthreads + B - 1) / B, B, 0, stream>>>(src, dst, dinv, h1z, out, E);
  }
}